// FactorizedSpectralConv2d_14121852469810
// MI455X (gfx1250) — compile-verified
//
#include <hip/hip_runtime.h>
#include <hip/hip_bf16.h>
#include <math.h>

typedef float v2f __attribute__((ext_vector_type(2)));
typedef float v8f __attribute__((ext_vector_type(8)));

#define Bn 8
#define Cn 64
#define On 64
#define Hn 256
#define Wn 256
#define HW (Hn * Wn)
#define MODES 16
#define PI_F 3.14159265358979323846f

// ---------------------------------------------------------------------------
// k1: per-(b,c) plane reductions. sumW[bc][h] = sum_w x, sumH[bc][w] = sum_h x
// 256 threads = 8 wave32s; wave `wv` walks rows h = wv, wv+8, ...; lane owns
// columns lane+32j. Row sums via wave shuffle reduce; col sums via LDS.
// Default RT loads: installs x into the 192MB L2 so k3 can re-read from L2.
// ---------------------------------------------------------------------------
__global__ __launch_bounds__(256) void k1_reduce(const float* __restrict__ x,
                                                 float* __restrict__ sumW,
                                                 float* __restrict__ sumH) {
  const int tid = threadIdx.x;
  const int lane = tid & 31;
  const int wv = tid >> 5;
  const int bc = blockIdx.x;  // 0..B*C-1
  const float* p = x + (size_t)bc * HW;

  float col[8];
#pragma unroll
  for (int j = 0; j < 8; ++j) col[j] = 0.0f;

  __shared__ float cs[8][256];

  for (int h = wv; h < Hn; h += 8) {
    const float* row = p + h * Wn;
    float rs = 0.0f;
#pragma unroll
    for (int j = 0; j < 8; ++j) {
      float v = row[lane + 32 * j];
      col[j] += v;
      rs += v;
    }
#pragma unroll
    for (int off = 16; off > 0; off >>= 1) rs += __shfl_xor(rs, off, 32);
    if (lane == 0) sumW[bc * Hn + h] = rs;
  }

#pragma unroll
  for (int j = 0; j < 8; ++j) cs[wv][lane + 32 * j] = col[j];
  __syncthreads();
  float t = 0.0f;
#pragma unroll
  for (int w = 0; w < 8; ++w) t += cs[w][tid];
  sumH[bc * Wn + tid] = t;
}

// ---------------------------------------------------------------------------
// k2a: 16-mode forward DFT of the means (both branches).
// ft[bc][k] = (1/sqrt(N)) * (1/N) * sum_t sum[bc][t] * e^{-2pi i k t / N}
// gid layout: [branch(2)][bc(512)][k(16)] = 16384 threads
// ---------------------------------------------------------------------------
__global__ void k2_dft(const float* __restrict__ sumW, const float* __restrict__ sumH,
                       float* __restrict__ ftx_r, float* __restrict__ ftx_i,
                       float* __restrict__ fty_r, float* __restrict__ fty_i) {
  int gid = blockIdx.x * blockDim.x + threadIdx.x;
  int k = gid & 15;
  int bc = (gid >> 4) & (Bn * Cn - 1);
  int br = gid >> 13;
  const float* s = ((br == 0) ? sumW : sumH) + bc * 256;
  float sr = 0.0f, si = 0.0f;
  const float w = -2.0f * PI_F * (float)k / 256.0f;
  for (int t = 0; t < 256; ++t) {
    float sn, cz;
    __sincosf(w * (float)t, &sn, &cz);
    float v = s[t];
    sr += v * cz;
    si += v * sn;  // e^{-i theta}: sin(w*t) already carries the minus sign
  }
  const float scale = 1.0f / (256.0f * 16.0f);  // mean (1/256) * ortho (1/16)
  float* dr = (br == 0) ? ftx_r : fty_r;
  float* di = (br == 0) ? ftx_i : fty_i;
  dr[bc * MODES + k] = sr * scale;
  di[bc * MODES + k] = si * scale;
}

// ---------------------------------------------------------------------------
// k2b: complex channel mix per mode. box[b][o][k] = sum_c ft[b][c][k]*w[c][o][k]
// gid layout: [branch(2)][b(8)][o(64)][k(16)] = 16384 threads
// ---------------------------------------------------------------------------
__global__ void k2_mix_modes(const float* __restrict__ ftx_r, const float* __restrict__ ftx_i,
                             const float* __restrict__ fty_r, const float* __restrict__ fty_i,
                             const float* __restrict__ w_x_r, const float* __restrict__ w_x_i,
                             const float* __restrict__ w_y_r, const float* __restrict__ w_y_i,
                             float* __restrict__ box_r, float* __restrict__ box_i,
                             float* __restrict__ boy_r, float* __restrict__ boy_i) {
  int gid = blockIdx.x * blockDim.x + threadIdx.x;
  int k = gid & 15;
  int o = (gid >> 4) & 63;
  int b = (gid >> 10) & 7;
  int br = gid >> 13;
  const float* fr = br ? fty_r : ftx_r;
  const float* fi = br ? fty_i : ftx_i;
  const float* wr = br ? w_y_r : w_x_r;
  const float* wi = br ? w_y_i : w_x_i;
  float ar = 0.0f, ai = 0.0f;
  for (int c = 0; c < Cn; ++c) {
    float xr = fr[(b * Cn + c) * MODES + k];
    float xi = fi[(b * Cn + c) * MODES + k];
    float vr = wr[(c * On + o) * MODES + k];
    float vi = wi[(c * On + o) * MODES + k];
    ar += xr * vr - xi * vi;
    ai += xr * vi + xi * vr;
  }
  float* dr = br ? boy_r : box_r;
  float* di = br ? boy_i : box_i;
  dr[(b * On + o) * MODES + k] = ar;
  di[(b * On + o) * MODES + k] = ai;
}

// ---------------------------------------------------------------------------
// k2c: inverse series. px[bo][pos] = (1/16)*(Re box0 + sum_{k=1}^{15} 2*(Re*cos - Im*sin))
// gid layout: [branch(2)][bo(512)][pos(256)] = 262144 threads
// (irfft ignores Im of bin 0; modes >= 16 are zero.)
// ---------------------------------------------------------------------------
__global__ void k2_profiles(const float* __restrict__ box_r, const float* __restrict__ box_i,
                            const float* __restrict__ boy_r, const float* __restrict__ boy_i,
                            float* __restrict__ px, float* __restrict__ py) {
  int gid = blockIdx.x * blockDim.x + threadIdx.x;
  int pos = gid & 255;
  int bo = (gid >> 8) & (Bn * On - 1);
  int br = gid >> 17;
  const float* rr = br ? boy_r : box_r;
  const float* ri = br ? boy_i : box_i;
  float val = rr[bo * MODES + 0];
  const float w = 2.0f * PI_F * (float)pos / 256.0f;
#pragma unroll
  for (int k = 1; k < MODES; ++k) {
    float sn, cz;
    __sincosf(w * (float)k, &sn, &cz);
    val += 2.0f * (rr[bo * MODES + k] * cz - ri[bo * MODES + k] * sn);
  }
  val *= (1.0f / 16.0f);  // ortho irfft 1/sqrt(256)
  float* dst = br ? py : px;
  dst[bo * 256 + pos] = val;
}

// ---------------------------------------------------------------------------
// k3: fused 1x1 mix GEMM (WMMA f32 16x16x4) + px/py/bias epilogue.
// Wave owns one 16-pixel N-tile; loads the B panel (K=64 x N=16, 32 VGPRs)
// once with NT (last-use) hints, then per o-tile batch-loads the A panel
// (16 K-steps x 2 VGPRs) so the 16 WMMAs issue back-to-back after a single
// wait instead of one load+wait per WMMA. Stores are nontemporal so the
// 128MiB output stream does not evict the L2-resident x.
// A 16x4 f32 layout: VGPR0 = K0 (lanes 0-15) / K2 (16-31); VGPR1 = K1/K3.
// B 4x16 mirrored. C/D: VGPR r = row M=r (lanes 0-15) / M=r+8 (lanes 16-31).
// ---------------------------------------------------------------------------
__global__ __launch_bounds__(256) void k3_mix_wmma(const float* __restrict__ x,
                                                   const float* __restrict__ mix_w,
                                                   const float* __restrict__ mix_b,
                                                   const float* __restrict__ px,
                                                   const float* __restrict__ py,
                                                   float* __restrict__ out) {
  const int tid = threadIdx.x;
  const int lane = tid & 31;
  const int wv = tid >> 5;
  const int b = blockIdx.y;
  const int n_base = (blockIdx.x * 8 + wv) * 16;  // tile never crosses a row (16 | 256)
  const int h = n_base >> 8;
  const int w0 = n_base & 255;
  const int nl = lane & 15;
  const int hi = lane >> 4;  // lane half selects K offset

  const float* xb = x + (size_t)b * Cn * HW;

  // B panel: 16 K-steps x 2 VGPRs, x read exactly once (NT: last use of x)
  v2f Bm[16];
#pragma unroll
  for (int kk = 0; kk < 16; ++kk) {
    int c0 = kk * 4;
    Bm[kk].x = __builtin_nontemporal_load(&xb[(size_t)(c0 + (hi ? 2 : 0)) * HW + n_base + nl]);
    Bm[kk].y = __builtin_nontemporal_load(&xb[(size_t)(c0 + (hi ? 3 : 1)) * HW + n_base + nl]);
  }

#pragma unroll
  for (int ot = 0; ot < 4; ++ot) {
    const int o0 = ot * 16;

    // Batched A panel for this o-tile: one load clause + one wait,
    // then 16 back-to-back WMMAs (only D->C accumulate dependency).
    v2f Aa[16];
#pragma unroll
    for (int kk = 0; kk < 16; ++kk) {
      const float* aw = mix_w + (o0 + nl) * Cn + kk * 4;
      Aa[kk].x = aw[hi ? 2 : 0];
      Aa[kk].y = aw[hi ? 3 : 1];
    }

    v8f acc = {};
#pragma unroll
    for (int kk = 0; kk < 16; ++kk) {
      acc = __builtin_amdgcn_wmma_f32_16x16x4_f32(false, Aa[kk], false, Bm[kk],
                                                  (short)0, acc, false, false);
    }

    // epilogue: out = acc + px[b,o,h] + py[b,o,w] + mix_b[o]
#pragma unroll
    for (int r = 0; r < 8; ++r) {
      int o = o0 + r + hi * 8;
      float v = acc[r] + px[((size_t)b * On + o) * Hn + h] +
                py[((size_t)b * On + o) * Wn + w0 + nl] + mix_b[o];
      __builtin_nontemporal_store(v, &out[((size_t)b * On + o) * HW + n_base + nl]);
    }
  }
}

// ---------------------------------------------------------------------------
extern "C" void kernel_launch(void* const* d_in, const int* in_sizes, int n_in,
                              void* d_out, int out_size, void* d_ws, size_t ws_size,
                              hipStream_t stream) {
  (void)in_sizes; (void)n_in; (void)out_size; (void)ws_size;
  const float* x     = (const float*)d_in[0];
  const float* w_x_r = (const float*)d_in[1];
  const float* w_x_i = (const float*)d_in[2];
  const float* w_y_r = (const float*)d_in[3];
  const float* w_y_i = (const float*)d_in[4];
  const float* mix_w = (const float*)d_in[5];
  const float* mix_b = (const float*)d_in[6];
  float* out = (float*)d_out;

  float* ws = (float*)d_ws;
  float* sumW  = ws;                    // [B*C*H]   131072
  float* sumH  = sumW + Bn * Cn * Hn;   // [B*C*W]   131072
  float* ftx_r = sumH + Bn * Cn * Wn;   // [B*C*16]  8192
  float* ftx_i = ftx_r + Bn * Cn * MODES;
  float* fty_r = ftx_i + Bn * Cn * MODES;
  float* fty_i = fty_r + Bn * Cn * MODES;
  float* box_r = fty_i + Bn * Cn * MODES;  // [B*O*16]
  float* box_i = box_r + Bn * On * MODES;
  float* boy_r = box_i + Bn * On * MODES;
  float* boy_i = boy_r + Bn * On * MODES;
  float* px    = boy_i + Bn * On * MODES;  // [B*O*H]
  float* py    = px + Bn * On * Hn;        // [B*O*W]

  k1_reduce<<<Bn * Cn, 256, 0, stream>>>(x, sumW, sumH);
  k2_dft<<<(2 * Bn * Cn * MODES) / 256, 256, 0, stream>>>(sumW, sumH, ftx_r, ftx_i,
                                                          fty_r, fty_i);
  k2_mix_modes<<<(2 * Bn * On * MODES) / 256, 256, 0, stream>>>(
      ftx_r, ftx_i, fty_r, fty_i, w_x_r, w_x_i, w_y_r, w_y_i,
      box_r, box_i, boy_r, boy_i);
  k2_profiles<<<(2 * Bn * On * 256) / 256, 256, 0, stream>>>(box_r, box_i, boy_r,
                                                             boy_i, px, py);
  dim3 g3(HW / (8 * 16), Bn);
  k3_mix_wmma<<<g3, 256, 0, stream>>>(x, mix_w, mix_b, px, py, out);
}